// GeneratorBA_77953656422787
// MI455X (gfx1250) — compile-verified
//
#include <hip/hip_runtime.h>
#include <hip/hip_bf16.h>

typedef float v2f __attribute__((ext_vector_type(2)));
typedef float v8f __attribute__((ext_vector_type(8)));

#define NNODES 512

// -------------------------------------------------------------------------
// d[b,c,i] = rsqrt(1 + sum_j adj[b,c,i,j])   (one wave per row)
// -------------------------------------------------------------------------
__global__ __launch_bounds__(256) void degree_kernel(
    const float* __restrict__ adj, float* __restrict__ dinv)
{
    const int tid  = threadIdx.x;
    const int wave = tid >> 5;
    const int lane = tid & 31;
    const size_t row = (size_t)blockIdx.x * 8 + wave; // over B*2*N rows
    const float* a = adj + row * NNODES;
    float s = 0.0f;
    for (int j = lane; j < NNODES; j += 32) s += a[j];
#pragma unroll
    for (int off = 16; off >= 1; off >>= 1) s += __shfl_xor(s, off, 32);
    if (lane == 0) dinv[row] = rsqrtf(s + 1.0f);
}

// -------------------------------------------------------------------------
// Y[bc] = act( Anorm[bc] (512x512) @ Z (512xF) + bias ),  Anorm fused on load.
// 8 waves/block; each wave owns TWO 16-row M-tiles (32 rows) x all F columns,
// so every B fragment (one ds_load_b64 from the LDS-staged, pair-interleaved
// Z slab) feeds two v_wmma_f32_16x16x4_f32.
// -------------------------------------------------------------------------
template<int NT, int F>
__global__ __launch_bounds__(256) void gcn_adj_gemm(
    const float* __restrict__ adj,
    const float* __restrict__ dinv,
    const float* __restrict__ Zp,
    const float* __restrict__ Zn,
    long zStride,
    const float* __restrict__ biasP,
    const float* __restrict__ biasN,
    float* __restrict__ out,
    int act)
{
    __shared__ float dcol[NNODES];
    __shared__ float zbuf[64 * NT * 16];   // 64-deep K slab, pair-interleaved

    const int bc  = blockIdx.y;
    const int c   = bc & 1;
    const int tid = threadIdx.x;
    const float* dRow = dinv + (size_t)bc * NNODES;
    for (int i = tid; i < NNODES; i += 256) dcol[i] = dRow[i];
    __syncthreads();

    const int wave = tid >> 5;
    const int lane = tid & 31;
    const int lm   = lane & 15;
    const int kh   = lane >> 4;
    const int tileM0 = blockIdx.x * 256 + wave * 32;   // two tiles: tileM0, tileM0+16
    const int m0 = tileM0 + lm;
    const int m1 = tileM0 + 16 + lm;
    const float dm0 = dcol[m0];
    const float dm1 = dcol[m1];
    const float* Arow0 = adj + ((size_t)bc * NNODES + m0) * NNODES;
    const float* Arow1 = adj + ((size_t)bc * NNODES + m1) * NNODES;
    const float* Z     = (c ? Zn : Zp) + (size_t)bc * zStride;
    const float* bias  = c ? biasN : biasP;

    v8f acc0[NT], acc1[NT];
#pragma unroll
    for (int t = 0; t < NT; ++t) { acc0[t] = v8f{}; acc1[t] = v8f{}; }

    for (int ks = 0; ks < NNODES; ks += 64) {
        __syncthreads();
        // stage Z rows [ks, ks+64) pair-interleaved: zbuf[(k/2)*F + col][2]
        for (int p = tid; p < 32 * F; p += 256) {
            const int kk  = (p / F) * 2;
            const int col = p - (p / F) * F;
            v2f zz;
            zz.x = Z[(size_t)(ks + kk) * F + col];
            zz.y = Z[(size_t)(ks + kk + 1) * F + col];
            *(v2f*)&zbuf[p * 2] = zz;
        }
        __syncthreads();

#pragma unroll 4
        for (int kq = 0; kq < 64; kq += 4) {
            const int ka = ks + kq + kh * 2;           // global k (even)
            float a0 = Arow0[ka], a1 = Arow0[ka + 1];
            float c0 = Arow1[ka], c1 = Arow1[ka + 1];
            a0 = (a0 + ((m0 == ka    ) ? 1.0f : 0.0f)) * dm0 * dcol[ka];
            a1 = (a1 + ((m0 == ka + 1) ? 1.0f : 0.0f)) * dm0 * dcol[ka + 1];
            c0 = (c0 + ((m1 == ka    ) ? 1.0f : 0.0f)) * dm1 * dcol[ka];
            c1 = (c1 + ((m1 == ka + 1) ? 1.0f : 0.0f)) * dm1 * dcol[ka + 1];
            v2f af0; af0.x = a0; af0.y = a1;
            v2f af1; af1.x = c0; af1.y = c1;
            const int pairBase = ((kq + kh * 2) >> 1) * F;  // pair index * F
#pragma unroll
            for (int t = 0; t < NT; ++t) {
                const int col = t * 16 + lm;
                v2f bfrag = v2f{};
                if (col < F) bfrag = *(const v2f*)&zbuf[(pairBase + col) * 2];
                acc0[t] = __builtin_amdgcn_wmma_f32_16x16x4_f32(
                    false, af0, false, bfrag, (short)0, acc0[t], false, false);
                acc1[t] = __builtin_amdgcn_wmma_f32_16x16x4_f32(
                    false, af1, false, bfrag, (short)0, acc1[t], false, false);
            }
        }
    }

    float* OutBC = out + (size_t)bc * NNODES * F;
#pragma unroll
    for (int t = 0; t < NT; ++t) {
        const int col = t * 16 + lm;
        if (col < F) {
            const float bv = bias[col];
#pragma unroll
            for (int v = 0; v < 8; ++v) {   // C layout: VGPRv -> rows v, v+8
                const int r0 = tileM0 + v + kh * 8;
                float v0 = acc0[t][v] + bv;
                float v1 = acc1[t][v] + bv;
                if (act) {
                    v0 = (v0 >= 0.0f) ? v0 : 0.2f * v0;
                    v1 = (v1 >= 0.0f) ? v1 : 0.2f * v1;
                }
                OutBC[(size_t)r0 * F + col]        = v0;
                OutBC[(size_t)(r0 + 16) * F + col] = v1;
            }
        }
    }
}

// -------------------------------------------------------------------------
// Z[bc] = X[bc] (512xK) @ W (KxF)    (per-branch weight select; W is tiny
// and L0/L2 resident, direct global fragment loads)
// -------------------------------------------------------------------------
template<int K, int F>
__global__ __launch_bounds__(256) void plain_gemm(
    const float* __restrict__ X,
    const float* __restrict__ Wp,
    const float* __restrict__ Wn,
    float* __restrict__ out)
{
    constexpr int NT = (F + 15) / 16;
    const int bc  = blockIdx.y;
    const int c   = bc & 1;
    const int tid = threadIdx.x;
    const int wave = tid >> 5, lane = tid & 31, lm = lane & 15, kh = lane >> 4;
    const int tileM = blockIdx.x * 128 + wave * 16;
    const int m     = tileM + lm;
    const float* Arow = X + ((size_t)bc * NNODES + m) * K;
    const float* W    = c ? Wn : Wp;

    v8f acc[NT];
#pragma unroll
    for (int t = 0; t < NT; ++t) acc[t] = v8f{};

#pragma unroll 4
    for (int k0 = 0; k0 < K; k0 += 4) {
        const int ka = k0 + kh * 2;
        v2f afrag; afrag.x = Arow[ka]; afrag.y = Arow[ka + 1];
#pragma unroll
        for (int t = 0; t < NT; ++t) {
            const int col = t * 16 + lm;
            v2f bfrag = v2f{};
            if (col < F) {
                bfrag.x = W[(size_t)ka * F + col];
                bfrag.y = W[(size_t)(ka + 1) * F + col];
            }
            acc[t] = __builtin_amdgcn_wmma_f32_16x16x4_f32(
                false, afrag, false, bfrag, (short)0, acc[t], false, false);
        }
    }

    float* OutBC = out + (size_t)bc * NNODES * F;
#pragma unroll
    for (int t = 0; t < NT; ++t) {
        const int col = t * 16 + lm;
        if (col < F) {
#pragma unroll
            for (int v = 0; v < 8; ++v)
                OutBC[(size_t)(tileM + v + kh * 8) * F + col] = acc[t][v];
        }
    }
}

// -------------------------------------------------------------------------
// Dynamic-conv head. One block per (b, branch). Reproduces the raw reshapes
// ((512,8)->(8,512) and the lp/ln interleave) with flat indexing, scattering
// straight into the final cc layout (B, 2ch, 512, 8).
// -------------------------------------------------------------------------
__global__ __launch_bounds__(256) void dcl_kernel(
    const float* __restrict__ conv3,  // (B*2, 512, 32)
    const float* __restrict__ conv4,  // (B*2, 512, 8)
    const float* __restrict__ wc,     // (72, 48)
    const int*  __restrict__ task,    // (B)
    float* __restrict__ cc)           // (B, 2, 512, 8)
{
    __shared__ float xc[48];
    __shared__ float part[8 * 32];
    __shared__ float wq[64];
    __shared__ float bq[8];
    const int bc = blockIdx.x;
    const int b  = bc >> 1, br = bc & 1;
    const int tid = threadIdx.x;
    const float* C3 = conv3 + (size_t)bc * NNODES * 32;
    const float* C4 = conv4 + (size_t)bc * NNODES * 8;

    { // mean over nodes
        const int col = tid & 31, grp = tid >> 5;
        float s = 0.0f;
        for (int r = grp * 64; r < grp * 64 + 64; ++r) s += C3[r * 32 + col];
        part[grp * 32 + col] = s;
    }
    __syncthreads();
    if (tid < 32) {
        float s = 0.0f;
        for (int g = 0; g < 8; ++g) s += part[g * 32 + tid];
        xc[tid] = s * (1.0f / NNODES);
    } else if (tid < 48) {
        xc[tid] = (task[b] == (tid - 32)) ? 1.0f : 0.0f;
    }
    __syncthreads();
    if (tid < 72) { // params = wc @ x_cond
        const float* wrow = wc + tid * 48;
        float s = 0.0f;
        for (int i = 0; i < 48; ++i) s += wrow[i] * xc[i];
        if (tid < 64) wq[tid] = s; else bq[tid - 64] = s;
    }
    __syncthreads();

    // logits[o,n] = sum_i wq[o,i] * head[i,n], head = raw-reshape of conv4
    float* ccB = cc + (size_t)b * 8192 + 8 * br;
    for (int q = tid; q < 4096; q += 256) { // q = o*512 + n  (also lp-flat index)
        const int o = q >> 9, n = q & 511;
        float s = bq[o];
#pragma unroll
        for (int i = 0; i < 8; ++i) s += wq[o * 8 + i] * C4[i * 512 + n];
        // lp[nn,oo] with nn=q>>3, oo=q&7 ; cc interleave: j = nn*16 + oo + 8*br
        ccB[(size_t)(q >> 3) * 16 + (q & 7)] = s;
    }
}

// -------------------------------------------------------------------------
// out[bch,i,j] = softplus( -sum_o cc[bch,i,o]*cc[bch,j,o] )
// cc slice (512x8, 16KB) staged in LDS; WMMA 16x16 tiles, K=8 (2 wmma).
// -------------------------------------------------------------------------
__global__ __launch_bounds__(256) void gram_softplus_kernel(
    const float* __restrict__ cc, float* __restrict__ out)
{
    __shared__ float Cs[NNODES * 8];
    const int bch = blockIdx.z;
    const int tid = threadIdx.x;
    const float* C = cc + (size_t)bch * NNODES * 8;
    for (int i = tid; i < NNODES * 8; i += 256) Cs[i] = C[i];
    __syncthreads();

    const int wave = tid >> 5, lane = tid & 31, lm = lane & 15, kh = lane >> 4;
    const int tileM = blockIdx.x * 64 + (wave >> 1) * 16;
    const int tileN = blockIdx.y * 32 + (wave & 1) * 16;

    v8f acc = v8f{};
#pragma unroll
    for (int k0 = 0; k0 < 8; k0 += 4) {
        const int kb = k0 + kh * 2;
        v2f a, bfr;
        a.x   = Cs[(tileM + lm) * 8 + kb];
        a.y   = Cs[(tileM + lm) * 8 + kb + 1];
        bfr.x = Cs[(tileN + lm) * 8 + kb];   // B[k,n] = C[n,k]
        bfr.y = Cs[(tileN + lm) * 8 + kb + 1];
        acc = __builtin_amdgcn_wmma_f32_16x16x4_f32(
            false, a, false, bfr, (short)0, acc, false, false);
    }

    float* O = out + (size_t)bch * NNODES * NNODES;
#pragma unroll
    for (int v = 0; v < 8; ++v) {
        const int row = tileM + v + kh * 8;
        const float y = -acc[v];                           // softplus(-x), stable
        const float r = fmaxf(y, 0.0f) + log1pf(expf(-fabsf(y)));
        O[(size_t)row * NNODES + tileN + lm] = r;
    }
}

// -------------------------------------------------------------------------
extern "C" void kernel_launch(void* const* d_in, const int* in_sizes, int n_in,
                              void* d_out, int out_size, void* d_ws, size_t ws_size,
                              hipStream_t stream)
{
    const float* adj = (const float*)d_in[0];
    const int*  task = (const int*) d_in[1];
    const float* w1p = (const float*)d_in[2];  const float* b1p = (const float*)d_in[3];
    const float* w2p = (const float*)d_in[4];  const float* b2p = (const float*)d_in[5];
    const float* w3p = (const float*)d_in[6];  const float* b3p = (const float*)d_in[7];
    const float* w4p = (const float*)d_in[8];  const float* b4p = (const float*)d_in[9];
    const float* w1n = (const float*)d_in[10]; const float* b1n = (const float*)d_in[11];
    const float* w2n = (const float*)d_in[12]; const float* b2n = (const float*)d_in[13];
    const float* w3n = (const float*)d_in[14]; const float* b3n = (const float*)d_in[15];
    const float* w4n = (const float*)d_in[16]; const float* b4n = (const float*)d_in[17];
    const float* wc  = (const float*)d_in[18];

    float* ws = (float*)d_ws;
    size_t off = 0;
    float* dinv = ws + off; off += (size_t)128 * NNODES;          // (B*2, 512)
    float* X1   = ws + off; off += (size_t)128 * NNODES * 128;
    float* Z2   = ws + off; off += (size_t)128 * NNODES * 64;
    float* X2   = ws + off; off += (size_t)128 * NNODES * 64;
    float* Z3   = ws + off; off += (size_t)128 * NNODES * 32;
    float* X3   = ws + off; off += (size_t)128 * NNODES * 32;
    float* Z4   = ws + off; off += (size_t)128 * NNODES * 8;
    float* X4   = ws + off; off += (size_t)128 * NNODES * 8;
    float* cc   = ws + off; off += (size_t)64 * 2 * NNODES * 8;

    dim3 blk(256);

    degree_kernel<<<dim3(128 * NNODES / 8), blk, 0, stream>>>(adj, dinv);

    // Layer 1: X1 = leaky(Anorm @ w1 + b1)   (Z = weights, shared -> stride 0)
    gcn_adj_gemm<8, 128><<<dim3(2, 128), blk, 0, stream>>>(
        adj, dinv, w1p, w1n, 0, b1p, b1n, X1, 1);
    // Layer 2
    plain_gemm<128, 64><<<dim3(4, 128), blk, 0, stream>>>(X1, w2p, w2n, Z2);
    gcn_adj_gemm<4, 64><<<dim3(2, 128), blk, 0, stream>>>(
        adj, dinv, Z2, Z2, (long)NNODES * 64, b2p, b2n, X2, 1);
    // Layer 3 (conv3)
    plain_gemm<64, 32><<<dim3(4, 128), blk, 0, stream>>>(X2, w3p, w3n, Z3);
    gcn_adj_gemm<2, 32><<<dim3(2, 128), blk, 0, stream>>>(
        adj, dinv, Z3, Z3, (long)NNODES * 32, b3p, b3n, X3, 1);
    // Layer 4 (conv4, no activation)
    plain_gemm<32, 8><<<dim3(4, 128), blk, 0, stream>>>(X3, w4p, w4n, Z4);
    gcn_adj_gemm<1, 8><<<dim3(2, 128), blk, 0, stream>>>(
        adj, dinv, Z4, Z4, (long)NNODES * 8, b4p, b4n, X4, 0);

    // Dynamic-conv head -> interleaved cc
    dcl_kernel<<<dim3(128), blk, 0, stream>>>(X3, X4, wc, task, cc);

    // Gram + softplus(-x) -> d_out (B,2,512,512)
    gram_softplus_kernel<<<dim3(8, 16, 128), blk, 0, stream>>>(cc, (float*)d_out);
}